// GATA_86440511799507
// MI455X (gfx1250) — compile-verified
//
#include <hip/hip_runtime.h>
#include <hip/hip_bf16.h>

typedef float v2f __attribute__((ext_vector_type(2)));
typedef float v8f __attribute__((ext_vector_type(8)));
typedef int   vi4 __attribute__((vector_size(16)));   // matches async-LDS builtin params

#if __has_builtin(__builtin_amdgcn_global_load_async_to_lds_b128)
#define HAVE_ASYNC_COPY 1
#else
#define HAVE_ASYNC_COPY 0
#endif

__device__ __forceinline__ void wait_async_all() {
#if __has_builtin(__builtin_amdgcn_s_wait_asynccnt)
    __builtin_amdgcn_s_wait_asynccnt(0);
#elif HAVE_ASYNC_COPY
    asm volatile("s_wait_asynccnt 0x0" ::: "memory");
#endif
}

// ---------------------------------------------------------------------------
// Generic f32 WMMA GEMM: C[M,N] (+)= act(A[M,K] @ op(B) + bias)
// One wave computes a 16x16 tile via V_WMMA_F32_16X16X4_F32.
// A rows strided by strideA (lets us gather pi[:, 1023, :] in place).
// TRANSB=1 means op(B)=B^T with B stored [N,K] row-major (ldb=K).
// ---------------------------------------------------------------------------
__device__ __forceinline__ float gelu_tanh(float x) {
    float u = 0.7978845608028654f * (x + 0.044715f * x * x * x);
    return 0.5f * x * (1.0f + tanhf(u));
}

template<bool TRANSB, bool HASBIAS, bool ACC, bool GELU>
__global__ __launch_bounds__(128)
void gemm_wmma_f32(const float* __restrict__ A, int strideA,
                   const float* __restrict__ B, int ldb,
                   const float* __restrict__ bias,
                   float* __restrict__ C, int ldc,
                   int M, int N, int K)
{
    const int lane = threadIdx.x & 31;
    const int wave = threadIdx.x >> 5;
    const int tile = blockIdx.x * (blockDim.x >> 5) + wave;
    const int tilesN = N >> 4;
    const int tm = tile / tilesN;
    const int tn = tile - tm * tilesN;
    if (tm * 16 >= M) return;
    const int m0 = tm << 4, n0 = tn << 4;
    const int half = lane >> 4;   // half 0 -> K {0,1}, half 1 -> K {2,3}
    const int lm = lane & 15;

    v8f acc = {};
    const float* Arow = A + (size_t)(m0 + lm) * strideA;
    for (int k = 0; k < K; k += 4) {
        const int kk = k + half * 2;
        v2f a, b;
        a.x = Arow[kk];
        a.y = Arow[kk + 1];
        if (TRANSB) {
            const float* Bc = B + (size_t)(n0 + lm) * ldb + kk;
            b.x = Bc[0];
            b.y = Bc[1];
        } else {
            const float* Br = B + (size_t)kk * ldb + (n0 + lm);
            b.x = Br[0];
            b.y = Br[ldb];
        }
        acc = __builtin_amdgcn_wmma_f32_16x16x4_f32(false, a, false, b,
                                                    (short)0, acc, false, false);
    }
    const int col = n0 + lm;
    const float bv = HASBIAS ? bias[col] : 0.0f;
#pragma unroll
    for (int r = 0; r < 8; ++r) {
        const int row = m0 + r + half * 8;
        float v = acc[r] + bv;
        if (GELU) v = gelu_tanh(v);
        float* cp = C + (size_t)row * ldc + col;
        if (ACC) v += *cp;
        *cp = v;
    }
}

// ---------------------------------------------------------------------------
// Single-pass streaming attention (single query per batch, online softmax).
//   scores_j = (rem_j . t_b + q_b . bk) / sqrt(D)
//   R[b]     = softmax(scores) @ rem     (v-projection deferred to a GEMM)
// rem is staged through LDS in 64-row tiles via async global->LDS copies, so
// HBM traffic is the 537 MB minimum (one pass), ~23 us at 23.3 TB/s.
// ---------------------------------------------------------------------------
#define TILE_ROWS 64
#define TILE_BYTES (TILE_ROWS * 512 * 4)

__global__ __launch_bounds__(256)
void attn_kernel(const float* __restrict__ pi,
                 const float* __restrict__ T,
                 const float* __restrict__ Q,
                 const float* __restrict__ bk,
                 float* __restrict__ R)
{
    extern __shared__ float tile[];          // TILE_ROWS x 512 (dynamic LDS)
    __shared__ float tvec[512];
    __shared__ float sc[TILE_ROWS];
    __shared__ float red[256];
    __shared__ float racc[8][512];

    const int b = blockIdx.x;
    const int tid = threadIdx.x;
    const int lane = tid & 31, wave = tid >> 5;
    const float* rem = pi + (size_t)b * (1024 * 512);

    tvec[tid]       = T[b * 512 + tid];
    tvec[tid + 256] = T[b * 512 + tid + 256];
    red[tid] = Q[b * 512 + tid] * bk[tid] + Q[b * 512 + tid + 256] * bk[tid + 256];
    __syncthreads();
    for (int s = 128; s > 0; s >>= 1) { if (tid < s) red[tid] += red[tid + s]; __syncthreads(); }
    const float c0 = red[0];
    __syncthreads();

    const float rsD = 0.044194173824159216f;   // 1/sqrt(512)
    float m = -1e30f;
    float Zpart = 0.f;
    float r0[16];
#pragma unroll
    for (int s = 0; s < 16; ++s) r0[s] = 0.f;

    for (int t0 = 0; t0 < 1023; t0 += TILE_ROWS) {
        const int rows = (1023 - t0 < TILE_ROWS) ? (1023 - t0) : TILE_ROWS;
        const int nchunk = rows * 128;          // 16B chunks in this tile
        {
            // interleaved 16B chunks: instruction i moves a contiguous 4 KB
            char* g = (char*)(size_t)(const char*)(rem + (size_t)t0 * 512);
            char* l = (char*)tile;
#if HAVE_ASYNC_COPY
#pragma unroll
            for (int i = 0; i < 32; ++i) {
                const int c = i * 256 + tid;
                if (c < nchunk) {
                    __builtin_amdgcn_global_load_async_to_lds_b128(
                        (vi4*)(g + (size_t)c * 16),
                        (vi4*)(l + c * 16),
                        0, 0);
                }
            }
            wait_async_all();
#else
#pragma unroll
            for (int i = 0; i < 32; ++i) {
                const int c = i * 256 + tid;
                if (c < nchunk) {
                    ((float4*)l)[c] = ((const float4*)g)[c];
                }
            }
#endif
        }
        __syncthreads();

        // scores for this tile (one wave per row)
        for (int jr = wave; jr < rows; jr += 8) {
            const float* rowp = tile + jr * 512;
            float d = 0.f;
#pragma unroll
            for (int s = 0; s < 16; ++s) d += rowp[lane + 32 * s] * tvec[lane + 32 * s];
#pragma unroll
            for (int mm = 16; mm > 0; mm >>= 1) d += __shfl_xor(d, mm, 32);
            if (lane == 0) sc[jr] = (d + c0) * rsD;
        }
        __syncthreads();

        // tile max -> merged running max
        red[tid] = (tid < rows) ? sc[tid] : -1e30f;
        __syncthreads();
        for (int s = 128; s > 0; s >>= 1) { if (tid < s) red[tid] = fmaxf(red[tid], red[tid + s]); __syncthreads(); }
        const float mnew = fmaxf(m, red[0]);
        __syncthreads();

        const float scale = __expf(m - mnew);
        Zpart *= scale;
#pragma unroll
        for (int s = 0; s < 16; ++s) r0[s] *= scale;

        // accumulate softmax-weighted rows from LDS
        for (int jr = wave; jr < rows; jr += 8) {
            const float wv = __expf(sc[jr] - mnew);
            if (lane == 0) Zpart += wv;
            const float* rowp = tile + jr * 512;
#pragma unroll
            for (int s = 0; s < 16; ++s) r0[s] += wv * rowp[lane + 32 * s];
        }
        m = mnew;
        __syncthreads();        // tile reused next iteration
    }

    // reduce Z over all threads
    red[tid] = Zpart; __syncthreads();
    for (int s = 128; s > 0; s >>= 1) { if (tid < s) red[tid] += red[tid + s]; __syncthreads(); }
    const float Z = red[0];
    __syncthreads();

#pragma unroll
    for (int s = 0; s < 16; ++s) racc[wave][32 * s + lane] = r0[s];
    __syncthreads();
    const float inv = 1.0f / Z;
    for (int d = tid; d < 512; d += 256) {
        float s8 = 0.f;
#pragma unroll
        for (int w = 0; w < 8; ++w) s8 += racc[w][d];
        R[b * 512 + d] = s8 * inv;
    }
}

// ---------------------------------------------------------------------------
// LayerNorm (eps=1e-5, population var) + per-row centering, written
// COLUMN-MAJOR into G (G[d*256 + n] = Xc[n][d]) for the Jacobi stage.
// ---------------------------------------------------------------------------
__global__ __launch_bounds__(256)
void ln_center_kernel(const float* __restrict__ Y,
                      const float* __restrict__ gamma,
                      const float* __restrict__ beta,
                      float* __restrict__ G)
{
    const int n = blockIdx.x, tid = threadIdx.x;
    __shared__ float red[256];
    const float x0 = Y[n * 512 + tid];
    const float x1 = Y[n * 512 + tid + 256];
    red[tid] = x0 + x1; __syncthreads();
    for (int s = 128; s > 0; s >>= 1) { if (tid < s) red[tid] += red[tid + s]; __syncthreads(); }
    const float mu = red[0] * (1.0f / 512.0f); __syncthreads();
    red[tid] = (x0 - mu) * (x0 - mu) + (x1 - mu) * (x1 - mu); __syncthreads();
    for (int s = 128; s > 0; s >>= 1) { if (tid < s) red[tid] += red[tid + s]; __syncthreads(); }
    const float rstd = rsqrtf(red[0] * (1.0f / 512.0f) + 1e-5f);
    __syncthreads();
    const float g0 = (x0 - mu) * rstd * gamma[tid] + beta[tid];
    const float g1 = (x1 - mu) * rstd * gamma[tid + 256] + beta[tid + 256];
    red[tid] = g0 + g1; __syncthreads();
    for (int s = 128; s > 0; s >>= 1) { if (tid < s) red[tid] += red[tid + s]; __syncthreads(); }
    const float m2 = red[0] * (1.0f / 512.0f);
    G[(size_t)tid * 256 + n]         = g0 - m2;
    G[(size_t)(tid + 256) * 256 + n] = g1 - m2;
}

// ---------------------------------------------------------------------------
// One-sided Jacobi SVD on G (256 rows x 512 cols, column-major).
// 256 disjoint column pairs per round, 511 rounds per sweep (tournament).
// Converged G columns = Xc @ eigvecs(cov) = the output projections.
// ---------------------------------------------------------------------------
__global__ __launch_bounds__(1024)
void jacobi_kernel(float* __restrict__ G, int sweeps)
{
    const int tid = threadIdx.x;
    const int lane = tid & 31, wave = tid >> 5;
    __shared__ int top[256], bot[256], t2[256], b2[256];
    __shared__ float cs_s[256], sn_s[256];
    __shared__ float nrm[512];

    if (tid < 256) { top[tid] = 2 * tid; bot[tid] = 2 * tid + 1; }
    __syncthreads();

    for (int sw = 0; sw < sweeps; ++sw) {
        if (tid < 512) {
            const float* col = G + (size_t)tid * 256;
            float a = 0.f;
            for (int i = 0; i < 256; ++i) { const float g = col[i]; a += g * g; }
            nrm[tid] = a;
        }
        __syncthreads();
        for (int round = 0; round < 511; ++round) {
            for (int pr = wave; pr < 256; pr += 32) {
                int p = top[pr], q = bot[pr];
                if (p > q) { const int t = p; p = q; q = t; }
                const float* gp = G + (size_t)p * 256;
                const float* gq = G + (size_t)q * 256;
                float d = 0.f;
#pragma unroll
                for (int s = 0; s < 8; ++s) d += gp[lane + 32 * s] * gq[lane + 32 * s];
#pragma unroll
                for (int mm = 16; mm > 0; mm >>= 1) d += __shfl_xor(d, mm, 32);
                if (lane == 0) {
                    const float a = nrm[p], bb = nrm[q];
                    float c = 1.f, s = 0.f;
                    if (fabsf(d) > 1e-20f + 1e-9f * sqrtf(a * bb)) {
                        const float zeta = (bb - a) / (2.f * d);
                        const float t = ((zeta >= 0.f) ? 1.f : -1.f) /
                                        (fabsf(zeta) + sqrtf(1.f + zeta * zeta));
                        c = rsqrtf(1.f + t * t);
                        s = t * c;
                        nrm[p] = a - t * d;
                        nrm[q] = bb + t * d;
                    }
                    cs_s[pr] = c; sn_s[pr] = s;
                }
            }
            __syncthreads();
            for (int pr = wave; pr < 256; pr += 32) {
                const float c = cs_s[pr], s = sn_s[pr];
                if (s != 0.f) {
                    int p = top[pr], q = bot[pr];
                    if (p > q) { const int t = p; p = q; q = t; }
                    float* gp = G + (size_t)p * 256;
                    float* gq = G + (size_t)q * 256;
#pragma unroll
                    for (int r = 0; r < 8; ++r) {
                        const int i = lane + 32 * r;
                        const float xp = gp[i], xq = gq[i];
                        gp[i] = c * xp - s * xq;
                        gq[i] = s * xp + c * xq;
                    }
                }
            }
            __syncthreads();
            if (tid < 256) {
                t2[tid] = (tid == 0) ? top[0] : ((tid == 1) ? bot[0] : top[tid - 1]);
                b2[tid] = (tid < 255) ? bot[tid + 1] : top[255];
            }
            __syncthreads();
            if (tid < 256) { top[tid] = t2[tid]; bot[tid] = b2[tid]; }
            __syncthreads();
        }
    }
}

// ---------------------------------------------------------------------------
// Column norms + bitonic sort (descending, index-descending tie-break),
// then gather the top-256 columns as the output projections.
// ---------------------------------------------------------------------------
__global__ __launch_bounds__(512)
void sort_kernel(const float* __restrict__ G, int* __restrict__ sidx)
{
    __shared__ float key[512];
    __shared__ int   idx[512];
    const int tid = threadIdx.x;
    const float* col = G + (size_t)tid * 256;
    float a = 0.f;
    for (int i = 0; i < 256; ++i) { const float g = col[i]; a += g * g; }
    key[tid] = a; idx[tid] = tid;
    __syncthreads();
    for (int k = 2; k <= 512; k <<= 1) {
        for (int j = k >> 1; j > 0; j >>= 1) {
            const int ixj = tid ^ j;
            if (ixj > tid) {
                const float ka = key[tid], kb = key[ixj];
                const int   ia = idx[tid], ib = idx[ixj];
                const bool aLess = (ka < kb) || (ka == kb && ia < ib);
                const bool doSwap = ((tid & k) == 0) ? aLess : !aLess;
                if (doSwap) { key[tid] = kb; key[ixj] = ka; idx[tid] = ib; idx[ixj] = ia; }
            }
            __syncthreads();
        }
    }
    if (tid < 256) sidx[tid] = idx[tid];
}

__global__ __launch_bounds__(256)
void gather_kernel(const float* __restrict__ G, const int* __restrict__ sidx,
                   float* __restrict__ out)
{
    const int k = blockIdx.x, n = threadIdx.x;
    const int id = sidx[k];
    out[(size_t)n * 256 + k] = G[(size_t)id * 256 + n];
}

// ---------------------------------------------------------------------------
extern "C" void kernel_launch(void* const* d_in, const int* in_sizes, int n_in,
                              void* d_out, int out_size, void* d_ws, size_t ws_size,
                              hipStream_t stream)
{
    (void)in_sizes; (void)n_in; (void)out_size; (void)ws_size;
    const float* pi    = (const float*)d_in[0];
    const float* Wq    = (const float*)d_in[1];
    const float* bq    = (const float*)d_in[2];
    const float* Wk    = (const float*)d_in[3];
    const float* bk    = (const float*)d_in[4];
    const float* Wv    = (const float*)d_in[5];
    const float* bv    = (const float*)d_in[6];
    const float* Wf    = (const float*)d_in[7];
    const float* bf    = (const float*)d_in[8];
    const float* W1    = (const float*)d_in[9];
    const float* b1    = (const float*)d_in[10];
    const float* W2    = (const float*)d_in[11];
    const float* b2    = (const float*)d_in[12];
    const float* gamma = (const float*)d_in[13];
    const float* beta  = (const float*)d_in[14];
    float* out = (float*)d_out;

    float* w  = (float*)d_ws;
    float* Q   = w;                    // 256x512
    float* T   = w + 131072;           // 256x512
    float* R   = w + 262144;           // 256x512
    float* Att = w + 393216;           // 256x512
    float* F   = w + 524288;           // 256x512
    float* Hb  = w + 655360;           // 256x2048
    float* Y   = w + 1179648;          // 256x512
    float* G   = w + 1310720;          // 512x256 (col-major Xc)
    int*   sidx = (int*)(w + 1441792); // 512

    const float* Xg = pi + 1023 * 512;        // pi[:, -1, :], row stride 1024*512
    const int strideXg = 1024 * 512;

    // allow >64KB dynamic LDS for the attention tile (idempotent, capture-safe)
    (void)hipFuncSetAttribute((const void*)attn_kernel,
                              hipFuncAttributeMaxDynamicSharedMemorySize,
                              TILE_BYTES);

    // Q = Xg @ Wq + bq
    gemm_wmma_f32<false, true, false, false><<<128, 128, 0, stream>>>(
        Xg, strideXg, Wq, 512, bq, Q, 512, 256, 512, 512);
    // T = Q @ Wk^T   (folds the k-projection into one per-batch vector)
    gemm_wmma_f32<true, false, false, false><<<128, 128, 0, stream>>>(
        Q, 512, Wk, 512, nullptr, T, 512, 256, 512, 512);
    // single-pass softmax-weighted average of rem rows
    attn_kernel<<<256, 256, TILE_BYTES, stream>>>(pi, T, Q, bk, R);
    // Att = R @ Wv + bv   (deferred v-projection)
    gemm_wmma_f32<false, true, false, false><<<128, 128, 0, stream>>>(
        R, 512, Wv, 512, bv, Att, 512, 256, 512, 512);
    // F = Xg @ Wf[:512] + bf ; F += Att @ Wf[512:]
    gemm_wmma_f32<false, true, false, false><<<128, 128, 0, stream>>>(
        Xg, strideXg, Wf, 512, bf, F, 512, 256, 512, 512);
    gemm_wmma_f32<false, false, true, false><<<128, 128, 0, stream>>>(
        Att, 512, Wf + 512 * 512, 512, nullptr, F, 512, 256, 512, 512);
    // FFN
    gemm_wmma_f32<false, true, false, true><<<512, 128, 0, stream>>>(
        F, 512, W1, 2048, b1, Hb, 2048, 256, 2048, 512);
    gemm_wmma_f32<false, true, false, false><<<128, 128, 0, stream>>>(
        Hb, 2048, W2, 512, b2, Y, 512, 256, 512, 2048);
    // LayerNorm + center -> G (column-major Xc)
    ln_center_kernel<<<256, 256, 0, stream>>>(Y, gamma, beta, G);
    // One-sided Jacobi: converged columns = Xc @ eigvecs
    jacobi_kernel<<<1, 1024, 0, stream>>>(G, 10);
    // rank columns by singular value, emit top-256 projections
    sort_kernel<<<1, 512, 0, stream>>>(G, sidx);
    gather_kernel<<<256, 256, 0, stream>>>(G, sidx, out);
}